// ElectrostaticEnergy_24945170055213
// MI455X (gfx1250) — compile-verified
//
#include <hip/hip_runtime.h>
#include <math.h>
#include <stdint.h>

#define THREADS   256
#define NBINS     2048
#define MAXBLOCKS 896

typedef float f32x4 __attribute__((ext_vector_type(4)));
typedef int   i32x4 __attribute__((ext_vector_type(4)));

__device__ __forceinline__ float fast_rcp(float x)  { return __builtin_amdgcn_rcpf(x); }
__device__ __forceinline__ float fast_sqrt(float x) { return __builtin_amdgcn_sqrtf(x); }
__device__ __forceinline__ float fast_rsq(float x)  { return __builtin_amdgcn_rsqf(x); }

// Async LDS->global 16B copy (CDNA5). Generic LDS pointers carry the wave-relative
// LDS byte offset in their low 32 bits (ISA 10.2), so a truncated generic pointer
// is a valid DS address for the async engine.
__device__ __forceinline__ void async_store_lds_b128(float* gdst, uint32_t lds_byte_addr)
{
#if defined(__HIP_DEVICE_COMPILE__) && defined(__gfx1250__)
  asm volatile("global_store_async_from_lds_b128 %0, %1, off"
               :: "v"((unsigned long long)(uintptr_t)gdst), "v"(lds_byte_addr)
               : "memory");
#else
  *(f32x4*)gdst = *(const f32x4*)(uintptr_t)
      ((uintptr_t)gdst ^ (uintptr_t)gdst ^ (uintptr_t)0);  // unreachable on host pass
#endif
}

__device__ __forceinline__ void wait_asynccnt0()
{
#if defined(__HIP_DEVICE_COMPILE__) && defined(__gfx1250__)
  asm volatile("s_wait_asynccnt 0" ::: "memory");
#endif
}

// Per-edge energy, matching the SpookyNet lr_cutoff electrostatics reference.
__device__ __forceinline__ float edge_energy(float rx, float ry, float rz,
                                             float qi, float qj)
{
  constexpr float KEHALF  = (float)(14.399645351950548 * 0.5);
  constexpr float CUTON16 = 2328306.4365386963f;      // 2.5^16
  constexpr float CUT_RC  = 0.009999999997526174f;    // 10^15/(10^16+2.5^16)^(17/16)
  constexpr float CUT_C   = 0.10999999999607098f;     // (10^16+2.5^16)^(-1/16)+CUT_RC

  const float d2  = fmaf(rx, rx, fmaf(ry, ry, rz * rz));
  const float d   = fast_sqrt(d2);
  const float fac = KEHALF * qi * qj;

  // switch f = fp/(fp+fm) = 1/(1 + exp(1/t - 1/(1-t))), t = (7.5-d)/5
  const float t  = (7.5f - d) * 0.2f;
  const float zz = fast_rcp(t) - fast_rcp(1.0f - t);
  float f = fast_rcp(1.0f + __expf(zz));            // inf-safe: exp(inf)->inf, rcp->0
  f = (t <= 0.0f) ? 0.0f : ((t >= 1.0f) ? 1.0f : f);
  const float omf = 1.0f - f;

  // long-range coulomb branch: 1/d + d/100 - 0.2, zero beyond d >= 10
  float coul = fmaf(d, 0.01f, fast_rcp(d)) - 0.2f;
  coul = (d < 10.0f) ? coul : 0.0f;

  // damped: (d^16 + 2.5^16)^(-1/16) + (1-f)*CUT_RC*d - CUT_C
  const float p2 = d * d, p4 = p2 * p2, p8 = p4 * p4, p16 = p8 * p8;
  const float x  = p16 + CUTON16;
  const float pw = fast_rsq(fast_sqrt(fast_sqrt(fast_sqrt(x))));   // x^(-1/16)
  const float damped = fmaf(omf * CUT_RC, d, pw) - CUT_C;

  return fac * fmaf(f, damped, omf * coul);
}

__global__ void es_zero_kernel(float* __restrict__ out, int n)
{
  int i = blockIdx.x * THREADS + threadIdx.x;
  if (i < n) out[i] = 0.0f;
}

// USE_LDS: accumulate into per-block LDS bins (requires M <= NBINS)
// WS_FLUSH: write bin row to ws[blockIdx.x*M ..] via async LDS->global stores
//           (reduced later by es_reduce_kernel); else flush with global atomics.
template <bool USE_LDS, bool WS_FLUSH>
__global__ __launch_bounds__(THREADS)
void es_edge_kernel(const float* __restrict__ q,  const float* __restrict__ r,
                    const int*   __restrict__ ii, const int*   __restrict__ jj,
                    const int*   __restrict__ im, float* __restrict__ out,
                    float* __restrict__ ws, int P, int M)
{
  __shared__ float bins[NBINS];
  if (USE_LDS) {
    for (int b = threadIdx.x; b < NBINS; b += THREADS) bins[b] = 0.0f;
    __syncthreads();
  }

  const int full   = P >> 2;           // chunks of 4 edges with vector loads
  const int total  = (P + 3) >> 2;
  const int stride = (int)(gridDim.x * THREADS);

  for (int c = (int)(blockIdx.x * THREADS + threadIdx.x); c < total; c += stride) {
    if (c < full) {
      const size_t rb = (size_t)c * 12u;
      const f32x4* rp = (const f32x4*)(r + rb);
      // software prefetch for the next grid-stride iteration (global_prefetch_b8)
      __builtin_prefetch((const void*)(r  + rb + (size_t)stride * 12u), 0, 0);
      __builtin_prefetch((const void*)(ii + (size_t)(c + stride) * 4u), 0, 0);
      __builtin_prefetch((const void*)(jj + (size_t)(c + stride) * 4u), 0, 0);

      // non-temporal: the edge stream is read-once; keep L2 for the q/idx_m gathers
      f32x4 A  = __builtin_nontemporal_load(rp + 0);
      f32x4 B  = __builtin_nontemporal_load(rp + 1);
      f32x4 C  = __builtin_nontemporal_load(rp + 2);
      i32x4 vi = __builtin_nontemporal_load((const i32x4*)ii + c);
      i32x4 vj = __builtin_nontemporal_load((const i32x4*)jj + c);

      const float e0 = edge_energy(A.x, A.y, A.z, q[vi.x], q[vj.x]);
      const float e1 = edge_energy(A.w, B.x, B.y, q[vi.y], q[vj.y]);
      const float e2 = edge_energy(B.z, B.w, C.x, q[vi.z], q[vj.z]);
      const float e3 = edge_energy(C.y, C.z, C.w, q[vi.w], q[vj.w]);
      const int m0 = im[vi.x], m1 = im[vi.y], m2 = im[vi.z], m3 = im[vi.w];

      if (USE_LDS) {
        atomicAdd(&bins[m0], e0);
        atomicAdd(&bins[m1], e1);
        atomicAdd(&bins[m2], e2);
        atomicAdd(&bins[m3], e3);
      } else {
        atomicAdd(out + m0, e0);
        atomicAdd(out + m1, e1);
        atomicAdd(out + m2, e2);
        atomicAdd(out + m3, e3);
      }
    } else {
      for (int e = c * 4; e < P; ++e) {          // scalar tail (P % 4)
        const size_t eb = (size_t)e * 3u;
        const float ev = edge_energy(r[eb], r[eb + 1], r[eb + 2], q[ii[e]], q[jj[e]]);
        const int m0 = im[ii[e]];
        if (USE_LDS) atomicAdd(&bins[m0], ev);
        else         atomicAdd(out + m0, ev);
      }
    }
  }

  if (USE_LDS) {
    __syncthreads();
    if (WS_FLUSH) {
      float* wrow = ws + (size_t)blockIdx.x * (size_t)M;
      const int nslice = M >> 2;
      const uint32_t lds_base = (uint32_t)(uintptr_t)(const void*)&bins[0];
      for (int s = (int)threadIdx.x; s < nslice; s += THREADS) {
#if defined(__HIP_DEVICE_COMPILE__) && defined(__gfx1250__)
        async_store_lds_b128(wrow + 4 * s, lds_base + 16u * (uint32_t)s);
#else
        *(f32x4*)(wrow + 4 * s) = *(const f32x4*)&bins[4 * s];
#endif
      }
      for (int b = (M & ~3) + (int)threadIdx.x; b < M; b += THREADS)
        wrow[b] = bins[b];                       // remainder bins (M % 4)
      wait_asynccnt0();
    } else {
      for (int b = (int)threadIdx.x; b < M; b += THREADS) {
        const float v = bins[b];
        if (v != 0.0f) atomicAdd(out + b, v);
      }
    }
  }
}

// out[b] += sum_{k in segment} ws[k*M + b]; coalesced 16B-per-lane column sums.
__global__ __launch_bounds__(THREADS)
void es_reduce_kernel(const float* __restrict__ ws, float* __restrict__ out,
                      int M, int K, int kChunk)
{
  const int nCols = M >> 2;
  const int col = (int)(blockIdx.x * THREADS + threadIdx.x);
  const int k0 = (int)blockIdx.y * kChunk;
  const int k1 = (k0 + kChunk < K) ? (k0 + kChunk) : K;

  if (col < nCols) {
    f32x4 s = {0.0f, 0.0f, 0.0f, 0.0f};
    for (int k = k0; k < k1; ++k)
      s += __builtin_nontemporal_load((const f32x4*)(ws + (size_t)k * (size_t)M) + col);
    float* o = out + 4 * col;
    atomicAdd(o + 0, s.x); atomicAdd(o + 1, s.y);
    atomicAdd(o + 2, s.z); atomicAdd(o + 3, s.w);
  }
  if (blockIdx.x == 0 && (int)threadIdx.x < (M & 3)) {   // remainder bins
    const int b = (M & ~3) + (int)threadIdx.x;
    float s = 0.0f;
    for (int k = k0; k < k1; ++k) s += ws[(size_t)k * (size_t)M + b];
    atomicAdd(out + b, s);
  }
}

extern "C" void kernel_launch(void* const* d_in, const int* in_sizes, int n_in,
                              void* d_out, int out_size, void* d_ws, size_t ws_size,
                              hipStream_t stream)
{
  (void)n_in;
  // inputs: [0]=atomic_numbers(unused) [1]=q [2]=r_ij [3]=idx_i [4]=idx_j [5]=idx_m [6]=maxm
  const float* q  = (const float*)d_in[1];
  const float* r  = (const float*)d_in[2];
  const int*   ii = (const int*)d_in[3];
  const int*   jj = (const int*)d_in[4];
  const int*   im = (const int*)d_in[5];
  float* out = (float*)d_out;

  const int P = in_sizes[3];
  const int M = out_size;

  const int total = (P + 3) >> 2;
  int blocks = (total + THREADS - 1) / THREADS;
  if (blocks > MAXBLOCKS) blocks = MAXBLOCKS;
  if (blocks < 1) blocks = 1;

  const bool ldsOK = (M <= NBINS);
  const size_t wsNeed = (size_t)blocks * (size_t)M * sizeof(float);
  const bool wsOK = ldsOK && (M >= 4) && (ws_size >= wsNeed) && (blocks >= 8);

  es_zero_kernel<<<(M + THREADS - 1) / THREADS, THREADS, 0, stream>>>(out, M);

  if (wsOK) {
    float* ws = (float*)d_ws;
    es_edge_kernel<true, true><<<blocks, THREADS, 0, stream>>>(q, r, ii, jj, im,
                                                               out, ws, P, M);
    const int nCols = M >> 2;
    int nseg = 16384 / (nCols > 0 ? nCols : 1);
    if (nseg > blocks) nseg = blocks;
    if (nseg < 1) nseg = 1;
    const int kChunk = (blocks + nseg - 1) / nseg;
    dim3 rgrid((unsigned)((nCols + THREADS - 1) / THREADS), (unsigned)nseg, 1u);
    es_reduce_kernel<<<rgrid, THREADS, 0, stream>>>(ws, out, M, blocks, kChunk);
  } else if (ldsOK) {
    es_edge_kernel<true, false><<<blocks, THREADS, 0, stream>>>(q, r, ii, jj, im,
                                                                out, nullptr, P, M);
  } else {
    es_edge_kernel<false, false><<<blocks, THREADS, 0, stream>>>(q, r, ii, jj, im,
                                                                 out, nullptr, P, M);
  }
}